// PatchInferer_31920196944414
// MI455X (gfx1250) — compile-verified
//
#include <hip/hip_runtime.h>
#include <stdint.h>

// Problem constants (match reference)
#define PW    0.5f
#define B_    2
#define C_    4
#define H_    128
#define N_    128
#define P_    32

// Volume strides (elements)
#define VOL_H  (H_ * H_)            // 16384
#define VOL_C  (H_ * H_ * H_)       // 2097152
#define VOL_B  (C_ * VOL_C)         // 8388608

// ---------------------------------------------------------------------------
// Kernel 1: out = (1 - PW) * vol, vectorized float4 (global_load/store_b128).
// Fully overwrites d_out -> deterministic regardless of poison state.
// 16,777,216 floats = 4,194,304 float4; 256 thr/block -> 16384 blocks.
// ---------------------------------------------------------------------------
__global__ void __launch_bounds__(256)
patchinf_scale_kernel(const float4* __restrict__ vol4, float4* __restrict__ out4) {
    const int i = blockIdx.x * 256 + threadIdx.x;
    float4 v = vol4[i];
    float4 r;
    r.x = v.x * (1.0f - PW);
    r.y = v.y * (1.0f - PW);
    r.z = v.z * (1.0f - PW);
    r.w = v.w * (1.0f - PW);
    out4[i] = r;
}

// ---------------------------------------------------------------------------
// Kernel 2: scatter-add PW * patches into out.
// One block per (n, c, h_local) plane: 32x32 = 1024 contiguous floats (4 KB).
// Stage the plane into LDS with the CDNA5 async copy engine
// (global_load_async_to_lds_b128, ASYNCcnt-tracked), then issue coalesced
// global_atomic_add_f32 into the (L2-resident) output volume.
// Grid: N*C*P = 128*4*32 = 16384 blocks of 256 threads (8 waves, wave32).
// ---------------------------------------------------------------------------
__global__ void __launch_bounds__(256)
patchinf_scatter_kernel(const float* __restrict__ patches,
                        const int*   __restrict__ centers,
                        float*       __restrict__ out) {
    __shared__ float tile[P_ * P_];          // 1024 floats = 4 KB

    const int t   = threadIdx.x;             // 0..255
    const int blk = blockIdx.x;              // 0..16383
    const int n   = blk / (C_ * P_);         // patch id
    const int rm  = blk % (C_ * P_);
    const int c   = rm / P_;                 // channel
    const int hl  = rm % P_;                 // local h within the patch

    // --- async copy: 256 lanes x 16 B = 4 KB plane into LDS --------------
    // GVS addressing: global = SADDR(u64) + VADDR(i32). Patch buffer is
    // 64 MB, so the 32-bit byte offset is safe.
    const unsigned src_off = (unsigned)((((n * C_ + c) * P_ + hl) * (P_ * P_)
                                         + t * 4) * (int)sizeof(float));
    // Low 32 bits of a flat shared pointer == LDS byte offset on gfx1250.
    const unsigned lds_off = (unsigned)(uintptr_t)(&tile[0]) + (unsigned)(t * 16);
    const unsigned long long sbase = (unsigned long long)(uintptr_t)patches;

    asm volatile("global_load_async_to_lds_b128 %0, %1, %2"
                 :
                 : "v"(lds_off), "v"(src_off), "s"(sbase)
                 : "memory");

    // --- overlap: uniform crop-center fetch (scalar loads) while copy runs
    const int b    = n / (N_ / B_);
    const int lo_h = centers[n * 3 + 0] - P_ / 2;
    const int lo_w = centers[n * 3 + 1] - P_ / 2;
    const int lo_d = centers[n * 3 + 2] - P_ / 2;
    // Reference guarantees lo in [0, H-P], so the 32^3 slice is in-bounds.
    const int base = b * VOL_B + c * VOL_C + (lo_h + hl) * VOL_H
                   + lo_w * H_ + lo_d;

    // --- wait for async copy, make LDS visible to all 8 waves ------------
    asm volatile("s_wait_asynccnt 0" ::: "memory");
    __syncthreads();

    // --- coalesced atomic scatter: 4 elements/thread ---------------------
    // e = t + k*256 covers the 1024-element plane; d (e & 31) contiguous.
#pragma unroll
    for (int k = 0; k < 4; ++k) {
        const int e  = t + k * 256;
        const int wl = e >> 5;               // local w
        const int dl = e & 31;               // local d
        atomicAdd(out + base + wl * H_ + dl, tile[e] * PW);
    }
}

// ---------------------------------------------------------------------------
// Launch wrapper
//   d_in[0] = patches      [128, 4, 32, 32, 32] f32
//   d_in[1] = vol          [2, 4, 128, 128, 128] f32
//   d_in[2] = crop_centers [128, 3] i32
//   d_out   = new_vol      [2, 4, 128, 128, 128] f32
// ---------------------------------------------------------------------------
extern "C" void kernel_launch(void* const* d_in, const int* in_sizes, int n_in,
                              void* d_out, int out_size, void* d_ws, size_t ws_size,
                              hipStream_t stream) {
    const float* patches = (const float*)d_in[0];
    const float* vol     = (const float*)d_in[1];
    const int*   centers = (const int*)d_in[2];
    float*       out     = (float*)d_out;

    // Kernel 1: scale 16.7M floats as 4.19M float4
    const int n4      = (B_ * C_ * H_ * H_ * H_) / 4;   // 4,194,304
    const int blocks1 = n4 / 256;                        // 16384
    patchinf_scale_kernel<<<blocks1, 256, 0, stream>>>(
        (const float4*)vol, (float4*)out);

    // Kernel 2: one block per (n, c, h) plane
    const int blocks2 = N_ * C_ * P_;                    // 16384
    patchinf_scatter_kernel<<<blocks2, 256, 0, stream>>>(patches, centers, out);
}